// PolynomialFilter_50190987821106
// MI455X (gfx1250) — compile-verified
//
#include <hip/hip_runtime.h>

// Polynomial graph filter: out = (sum_i c_i * L^i) @ x, N=4096, D=256, K=8.
// Horner on the vector block: y = c7*x; y = L@y + c_k*x for k=6..0.
// 7 dependent (4096x4096)@(4096x256) GEMMs (~60 GFLOP total) instead of 7
// N^3 matmuls (~960 GFLOP). L (64 MB fp32) lives in the 192 MB L2 after the
// first pass. fp32 WMMA (v_wmma_f32_16x16x4_f32) keeps full precision.

typedef __attribute__((ext_vector_type(2))) float v2f;
typedef __attribute__((ext_vector_type(8))) float v8f;

#define N_DIM 4096
#define D_DIM 256
#define BM 64
#define BN 64
#define KT 16
#define A_STRIDE 20    // A tile row stride (floats): conflict-free, 16B-aligned rows
#define BPS 160        // B k-pair stride (floats): 160 % 64 == 32 -> half-wave bank split
#define A_BUF (BM * A_STRIDE)        // 1280 floats per buffer
#define B_BUF ((KT / 2) * BPS)       // 1280 floats per buffer

__global__ __launch_bounds__(256) void poly_init_kernel(const float* __restrict__ x,
                                                        const float* __restrict__ coeffs,
                                                        float* __restrict__ y) {
    int i = blockIdx.x * blockDim.x + threadIdx.x;
    y[i] = coeffs[7] * x[i];
}

__global__ __launch_bounds__(256) void poly_gemm_kernel(const float* __restrict__ L,
                                                        const float* __restrict__ Y,
                                                        const float* __restrict__ X,
                                                        const float* __restrict__ coeffs,
                                                        int kIdx,
                                                        float* __restrict__ Out) {
    __shared__ float sA[2 * A_BUF];  // [buf][row 0..63][k 0..15], stride 20
    __shared__ float sB[2 * B_BUF];  // [buf][kpair 0..7][n 0..63][k&1], stride 160

    const int t    = threadIdx.x;
    const int lane = t & 31;          // wave32
    const int wave = t >> 5;          // 8 waves
    const int half = lane >> 4;
    const int l16  = lane & 15;

    // 8 waves tile the 64x64 block: 4 along M (16 rows), 2 along N (32 cols).
    const int mWave  = (wave & 3) * 16;
    const int nWave  = (wave >> 2) * 32;
    const int mBlock = blockIdx.x * BM;
    const int nBlock = blockIdx.y * BN;

    // Cooperative float4 fill assignments (256 threads).
    const int ar = t >> 2;            // 0..63    A row
    const int ac = (t & 3) << 2;      // 0,4,8,12 A col
    const int br = t >> 4;            // 0..15    B (k) row
    const int bc = (t & 15) << 2;     // 0..60    B (n) col
    const int bkp = br >> 1;          // k-pair
    const int bko = br & 1;           // k parity -> interleave slot

    const long aRowBase = (long)(mBlock + ar) * N_DIM + ac;

    v8f acc0 = {};                    // (mWave, nWave +  0) 16x16 tile
    v8f acc1 = {};                    // (mWave, nWave + 16) 16x16 tile

    // Prologue: stage tile 0 into buffer 0.
    float4 ra = *(const float4*)&L[aRowBase];
    float4 rb = *(const float4*)&Y[br * D_DIM + nBlock + bc];
    *(float4*)&sA[ar * A_STRIDE + ac] = ra;
    {
        float* p = &sB[bkp * BPS + 2 * bc + bko];
        p[0] = rb.x; p[2] = rb.y; p[4] = rb.z; p[6] = rb.w;
    }

    int cur = 0;
    for (int kk = 0; kk < N_DIM; kk += KT) {
        __syncthreads();              // buffer `cur` fully written, safe to read

        const bool more = (kk + KT) < N_DIM;
        if (more) {                   // stage next tile into registers (overlaps WMMA)
            ra = *(const float4*)&L[aRowBase + kk + KT];
            rb = *(const float4*)&Y[(kk + KT + br) * D_DIM + nBlock + bc];
            __builtin_prefetch(&L[aRowBase + kk + 2 * KT], 0, 3);
        }

        const int aOff = cur * A_BUF;
        const int bOff = cur * B_BUF;
#pragma unroll
        for (int kb = 0; kb < KT / 4; ++kb) {
            const int kp = 2 * kb + half;   // k-pair index: krow = 4*kb + 2*half
            // A fragment (16x4): lane = M, half-wave picks K pair -> one ds_load_b64.
            v2f a = *(const v2f*)&sA[aOff + (mWave + l16) * A_STRIDE + 4 * kb + 2 * half];
            // B fragments (4x16): pair-interleaved -> single aligned ds_load_b64 each.
            v2f b0 = *(const v2f*)&sB[bOff + kp * BPS + 2 * (nWave + l16)];
            v2f b1 = *(const v2f*)&sB[bOff + kp * BPS + 2 * (nWave + 16 + l16)];
            acc0 = __builtin_amdgcn_wmma_f32_16x16x4_f32(
                false, a, false, b0, (short)0, acc0, false, false);
            acc1 = __builtin_amdgcn_wmma_f32_16x16x4_f32(
                false, a, false, b1, (short)0, acc1, false, false);
        }

        if (more) {                   // write staged tile into the *other* buffer:
            const int na = (cur ^ 1) * A_BUF;   // no hazard vs readers of `cur`
            const int nb = (cur ^ 1) * B_BUF;
            *(float4*)&sA[na + ar * A_STRIDE + ac] = ra;
            float* p = &sB[nb + bkp * BPS + 2 * bc + bko];
            p[0] = rb.x; p[2] = rb.y; p[4] = rb.z; p[6] = rb.w;
        }
        cur ^= 1;
    }

    // Epilogue: Out = acc + c * X. C layout: M = 8*half + vgpr, N = lane%16.
    const float c    = coeffs[kIdx];
    const int   mrow = mBlock + mWave + 8 * half;
    const int   ncol = nBlock + nWave + l16;
#pragma unroll
    for (int r = 0; r < 8; ++r) {
        int i0 = (mrow + r) * D_DIM + ncol;
        Out[i0]      = acc0[r] + c * X[i0];
        Out[i0 + 16] = acc1[r] + c * X[i0 + 16];
    }
}

extern "C" void kernel_launch(void* const* d_in, const int* in_sizes, int n_in,
                              void* d_out, int out_size, void* d_ws, size_t ws_size,
                              hipStream_t stream) {
    (void)in_sizes; (void)n_in; (void)out_size; (void)ws_size;
    const float* L      = (const float*)d_in[0];
    const float* x      = (const float*)d_in[1];
    const float* coeffs = (const float*)d_in[2];
    float* out = (float*)d_out;
    float* ws  = (float*)d_ws;   // needs N*D*4 = 4 MB

    // y = c7 * x  -> ws
    poly_init_kernel<<<(N_DIM * D_DIM) / 256, 256, 0, stream>>>(x, coeffs, ws);

    // Horner: 7 passes, ping-pong ws <-> out so the final pass (k=0) writes out.
    dim3 grid(N_DIM / BM, D_DIM / BN);
    const float* src = ws;
    for (int p = 0; p < 7; ++p) {
        int    kIdx = 6 - p;
        float* dst  = (p & 1) ? ws : out;
        poly_gemm_kernel<<<grid, 256, 0, stream>>>(L, src, x, coeffs, kIdx, dst);
        src = dst;
    }
}